// InferenceFeatureFusionRGBXYZ_23450521436163
// MI455X (gfx1250) — compile-verified
//
#include <hip/hip_runtime.h>
#include <math.h>

// ---------------------------------------------------------------------------
// Problem constants (from reference)
// ---------------------------------------------------------------------------
#define F    512          // feature dim
#define NL   1600         // left samples
#define NR   64           // right samples
#define MTOT (NL + NR)    // 1664 stacked GEMM rows
#define HID  512          // hidden width (W1 columns)

typedef float v2f __attribute__((ext_vector_type(2)));
typedef float v4f __attribute__((ext_vector_type(4)));
typedef float v8f __attribute__((ext_vector_type(8)));
typedef int   v4i __attribute__((vector_size(16)));   // matches async builtin param

#define AS1 __attribute__((address_space(1)))
#define AS3 __attribute__((address_space(3)))

#if __has_builtin(__builtin_amdgcn_global_load_async_to_lds_b128) && \
    __has_builtin(__builtin_amdgcn_s_wait_asynccnt)
#define USE_ASYNC_LDS 1
#else
#define USE_ASYNC_LDS 0
#endif

// ---------------------------------------------------------------------------
// Stage 1: [1664 x 512] = gather(rows) @ W1_half, with b1 folded into the
// 64 "right" rows.  BM=64 BN=64 BK=32, 8 waves per block, WMMA f32 16x16x4.
// A chunk staged via async global->LDS (ASYNCcnt); B chunk staged TRANSPOSED
// so every WMMA B-fragment is one aligned ds_load_b64 (no repack movs).
// ---------------------------------------------------------------------------
#define BM 64
#define BN 64
#define BK 32
#define LDA_S 36   // LDS row stride (floats) for A chunk [m][k]: 16B-aligned, bank-clean
#define LDBT  34   // LDS row stride (floats) for Bt chunk [n][k]: 8B-aligned pairs, bank-clean

__global__ __launch_bounds__(256)
void fusion_gemm_stage1(const float* __restrict__ soutput_l,
                        const float* __restrict__ soutput_r,
                        const float* __restrict__ W1,   // [2F][HID]
                        const float* __restrict__ b1,   // [HID]
                        const int*   __restrict__ idx_l,
                        const int*   __restrict__ idx_r,
                        float*       __restrict__ out)  // [MTOT][HID]
{
    __shared__ float As[BM * LDA_S];   // A chunk, row-major [m][k]
    __shared__ float Bt[BN * LDBT];    // B chunk, TRANSPOSED [n][k]

    const int tid = threadIdx.x;
    const int bm  = blockIdx.x;            // 0..25 (25 == the right-rows block)
    const int nb  = blockIdx.y * BN;       // 0..448 step 64
    const bool right = (bm == (NL / BM));  // 1600/64 == 25, exact split

    // ---- gather sources for A loads (2 rows per thread, fixed all K-chunks)
    const int row0 = tid >> 3;             // 0..31
    const int row1 = row0 + 32;            // 32..63
    const int c4a  = (tid & 7) * 4;        // float4 column within K-chunk
    const float* srcBase = right ? soutput_r : soutput_l;
    const int g0 = right ? idx_r[row0] : idx_l[bm * BM + row0];
    const int g1 = right ? idx_r[row1] : idx_l[bm * BM + row1];
    const float* aRow0 = srcBase + (size_t)g0 * F;
    const float* aRow1 = srcBase + (size_t)g1 * F;

    // ---- B (W1) loads: 2 rows per thread per K-chunk, stored transposed
    const int w1base = right ? F : 0;      // bottom half of W1 for right rows
    const int kr0 = tid >> 4;              // 0..15
    const int kr1 = kr0 + 16;              // 16..31
    const int c4b = (tid & 15) * 4;        // n within block tile

    // ---- wave tiling: 4 (M) x 2 (N) waves, each 16x32 -> two 16x16 tiles
    const int wave   = tid >> 5;
    const int lane   = tid & 31;
    const int lane16 = lane & 15;
    const int half   = lane >> 4;          // selects K pair {0,1} vs {2,3}
    const int wm = (wave >> 1) * 16;       // 0,16,32,48
    const int wn = (wave & 1) * 32;        // 0,32

    const float* aFrag  = &As[(wm + lane16) * LDA_S];
    const float* btRow0 = &Bt[(wn + lane16) * LDBT];
    const float* btRow1 = &Bt[(wn + 16 + lane16) * LDBT];

#if USE_ASYNC_LDS
    // exact builtin pointer types: AS1 v4i* source, AS3 v4i* LDS destination
    AS3 v4i* ldsA0 = (AS3 v4i*)((AS3 float*)&As[row0 * LDA_S + c4a]);
    AS3 v4i* ldsA1 = (AS3 v4i*)((AS3 float*)&As[row1 * LDA_S + c4a]);
#endif

    v8f acc0 = {};
    v8f acc1 = {};

    for (int kb = 0; kb < F; kb += BK) {
        // ---- A chunk: 64 rows x 32 cols (gathered), async straight to LDS
#if USE_ASYNC_LDS
        __builtin_amdgcn_global_load_async_to_lds_b128(
            (AS1 v4i*)(aRow0 + kb + c4a), ldsA0, 0, 0);
        __builtin_amdgcn_global_load_async_to_lds_b128(
            (AS1 v4i*)(aRow1 + kb + c4a), ldsA1, 0, 0);
#else
        *(v4f*)&As[row0 * LDA_S + c4a] = *(const v4f*)(aRow0 + kb + c4a);
        *(v4f*)&As[row1 * LDA_S + c4a] = *(const v4f*)(aRow1 + kb + c4a);
#endif
        // ---- B chunk: W1 rows [w1base+kb, +32), cols [nb, nb+64), transposed
        {
            v4f w0 = *(const v4f*)(W1 + (size_t)(w1base + kb + kr0) * HID + nb + c4b);
            v4f w1 = *(const v4f*)(W1 + (size_t)(w1base + kb + kr1) * HID + nb + c4b);
            Bt[(c4b + 0) * LDBT + kr0] = w0.x;
            Bt[(c4b + 1) * LDBT + kr0] = w0.y;
            Bt[(c4b + 2) * LDBT + kr0] = w0.z;
            Bt[(c4b + 3) * LDBT + kr0] = w0.w;
            Bt[(c4b + 0) * LDBT + kr1] = w1.x;
            Bt[(c4b + 1) * LDBT + kr1] = w1.y;
            Bt[(c4b + 2) * LDBT + kr1] = w1.z;
            Bt[(c4b + 3) * LDBT + kr1] = w1.w;
        }
#if USE_ASYNC_LDS
        __builtin_amdgcn_s_wait_asynccnt(0);
#endif
        __syncthreads();

#pragma unroll
        for (int ks = 0; ks < BK; ks += 4) {
            const int kk = ks + half * 2;
            // A frag: row M = wm+lane16, K pair per lane half (ISA 16x4 layout)
            v2f a = *(const v2f*)(aFrag + kk);
            // B frags: col N per lane, contiguous (kk, kk+1) from transposed tile
            v2f b0  = *(const v2f*)(btRow0 + kk);
            v2f b1f = *(const v2f*)(btRow1 + kk);
            acc0 = __builtin_amdgcn_wmma_f32_16x16x4_f32(
                false, a, false, b0,  (short)0, acc0, false, false);
            acc1 = __builtin_amdgcn_wmma_f32_16x16x4_f32(
                false, a, false, b1f, (short)0, acc1, false, false);
        }
        __syncthreads();
    }

    // D layout: VGPR i -> row (i + 8*half), col = lane16
    const int col0 = nb + wn + lane16;
    const int col1 = col0 + 16;
    const float bias0 = right ? b1[col0] : 0.0f;
    const float bias1 = right ? b1[col1] : 0.0f;
#pragma unroll
    for (int i = 0; i < 8; ++i) {
        const int grow = bm * BM + wm + half * 8 + i;
        out[(size_t)grow * HID + col0] = acc0[i] + bias0;
        out[(size_t)grow * HID + col1] = acc1[i] + bias1;
    }
}

// ---------------------------------------------------------------------------
// Stage 2: out[l][r] = 1/(1+exp(d)),
//   d = sum_h relu(A[l,h]+B'[r,h]) * (W2[h,0]-W2[h,1]) + (b2[0]-b2[1])
// One block per l; 256 threads = 64 r-values x 4 h-segments of 128.
// ---------------------------------------------------------------------------
__global__ __launch_bounds__(256)
void fusion_stage2(const float* __restrict__ AB,   // [MTOT][HID] from stage 1
                   const float* __restrict__ W2,   // [HID][2]
                   const float* __restrict__ b2,   // [2]
                   float*       __restrict__ out)  // [NL][NR]
{
    __shared__ float Arow[HID];
    __shared__ float wd[HID];
    __shared__ float partial[NR][4];

    const int tid = threadIdx.x;
    const int l   = blockIdx.x;

    if (tid < 128) {
        // A row for this l
        *(v4f*)&Arow[tid * 4] = *(const v4f*)(AB + (size_t)l * HID + tid * 4);
    } else {
        // weight-difference vector: wd[h] = W2[h][0] - W2[h][1]
        const int h4 = (tid - 128) * 4;
        v4f w0 = *(const v4f*)(W2 + h4 * 2);      // rows h4, h4+1
        v4f w1 = *(const v4f*)(W2 + h4 * 2 + 4);  // rows h4+2, h4+3
        wd[h4 + 0] = w0.x - w0.y;
        wd[h4 + 1] = w0.z - w0.w;
        wd[h4 + 2] = w1.x - w1.y;
        wd[h4 + 3] = w1.z - w1.w;
    }
    __syncthreads();

    const int r   = tid & 63;
    const int seg = tid >> 6;
    const float* Brow = AB + (size_t)(NL + r) * HID + seg * 128;  // b1 pre-folded
    const float* Ar   = Arow + seg * 128;
    const float* Wd   = wd + seg * 128;

    float acc = 0.0f;
#pragma unroll 8
    for (int h = 0; h < 128; h += 4) {
        v4f bv = *(const v4f*)(Brow + h);
        v4f av = *(const v4f*)(Ar + h);
        v4f wv = *(const v4f*)(Wd + h);
        acc += fmaxf(av.x + bv.x, 0.0f) * wv.x;
        acc += fmaxf(av.y + bv.y, 0.0f) * wv.y;
        acc += fmaxf(av.z + bv.z, 0.0f) * wv.z;
        acc += fmaxf(av.w + bv.w, 0.0f) * wv.w;
    }
    partial[r][seg] = acc;
    __syncthreads();

    if (tid < NR) {
        const float d = partial[tid][0] + partial[tid][1] +
                        partial[tid][2] + partial[tid][3] +
                        (b2[0] - b2[1]);
        out[(size_t)l * NR + tid] = 1.0f / (1.0f + __expf(d));
    }
}

// ---------------------------------------------------------------------------
// Launcher
// ---------------------------------------------------------------------------
extern "C" void kernel_launch(void* const* d_in, const int* in_sizes, int n_in,
                              void* d_out, int out_size, void* d_ws, size_t ws_size,
                              hipStream_t stream)
{
    (void)in_sizes; (void)n_in; (void)out_size; (void)ws_size;
    const float* soutput_l = (const float*)d_in[0];
    const float* soutput_r = (const float*)d_in[1];
    const float* W1        = (const float*)d_in[2];
    const float* b1        = (const float*)d_in[3];
    const float* W2        = (const float*)d_in[4];
    const float* b2        = (const float*)d_in[5];
    const int*   idx_l     = (const int*)d_in[6];
    const int*   idx_r     = (const int*)d_in[7];

    float* ws  = (float*)d_ws;    // needs MTOT*HID*4 = 3.4 MB
    float* out = (float*)d_out;   // [1600][64]

    dim3 g1(MTOT / BM, HID / BN);           // (26, 8)
    fusion_gemm_stage1<<<g1, 256, 0, stream>>>(soutput_l, soutput_r, W1, b1,
                                               idx_l, idx_r, ws);
    fusion_stage2<<<NL, 256, 0, stream>>>(ws, W2, b2, out);
}